// ScaledDotProductAttention_2_31834297598141
// MI455X (gfx1250) — compile-verified
//
#include <hip/hip_runtime.h>
#include <math.h>

typedef __attribute__((ext_vector_type(2))) float v2f;
typedef __attribute__((ext_vector_type(8))) float v8f;

#define SEQ    4096
#define DH     64
#define NH     8
#define NBATCH 2
#define SM_SCALE 0.125f       // 1/sqrt(64)
#define MASK_NEG (-1e9f)

// One wave (32 lanes) owns one 16-query tile of one (b,h).
// Pass 1: QK^T via V_WMMA_F32_16X16X4_F32 (two independent accumulator
//         chains to halve XDL RAW latency), per-lane online softmax (m, l);
//         mask tile compressed to bits via wave32 ballot -> LDS (8KB/wave).
// Merge (m,l) across the 16-lane halves with width-16 xor shuffles.
// Pass 2: recompute QK^T (cheaper than spilling 1GB of raw scores), mask
//         bits replayed from LDS, P = exp(s-m)/l -> nontemporal store,
//         P transposed through LDS into A-fragment layout, P@V via WMMA.
__global__ __launch_bounds__(256) void attn_fused_kernel(
    const float* __restrict__ Q, const float* __restrict__ K,
    const float* __restrict__ V, const int* __restrict__ MASK,
    float* __restrict__ CTX, float* __restrict__ ATT)
{
    __shared__ float    plds[8][16 * 17];        // per-wave 16x16 P tile, stride 17 -> conflict-free
    __shared__ unsigned mlds[8][(SEQ / 16) * 8]; // per-wave mask bits: 8 dwords per key tile

    const int lane = threadIdx.x & 31;
    const int wav  = threadIdx.x >> 5;
    const int half = lane >> 4;   // 0: lanes 0-15, 1: lanes 16-31
    const int l16  = lane & 15;

    const int w  = blockIdx.x * 8 + wav;   // global wave id: 0..4095
    const int bh = w >> 8;                 // 256 query tiles per (b,h)
    const int q0 = (w & 255) << 4;
    const int b  = bh >> 3;                // H == 8

    const float* qb = Q + (size_t)bh * SEQ * DH;
    const float* kb = K + (size_t)bh * SEQ * DH;
    const float* vb = V + (size_t)bh * SEQ * DH;
    const int*   mb = MASK + (size_t)b * SEQ * SEQ;
    float* attb = ATT + (size_t)bh * SEQ * SEQ;
    float* ctxb = CTX + (size_t)bh * SEQ * DH;

    unsigned* mwave = mlds[wav];

    // ---- Load Q tile as 16 A-fragments (16x4 f32 each): contiguous float2/lane
    v2f qa[16];
    {
        const float* qrow = qb + (q0 + l16) * DH + 2 * half;
        #pragma unroll
        for (int c = 0; c < 16; ++c)
            qa[c] = *(const v2f*)(qrow + 4 * c);
    }

    // ---- Pass 1: online softmax statistics + mask bit compression
    float mrow[8], lrow[8];
    #pragma unroll
    for (int i = 0; i < 8; ++i) { mrow[i] = -1e30f; lrow[i] = 0.0f; }

    for (int n0 = 0; n0 < SEQ; n0 += 16) {
        const float* krow = kb + (n0 + l16) * DH + 2 * half;
        __builtin_prefetch(krow + 16 * DH, 0, 1);   // next key tile -> global_prefetch_b8
        v8f acc0 = {}, acc1 = {};                   // two chains: halve WMMA RAW latency
        #pragma unroll
        for (int c = 0; c < 8; ++c) {
            v2f kf0 = *(const v2f*)(krow + 4 * (2 * c));
            v2f kf1 = *(const v2f*)(krow + 4 * (2 * c + 1));
            acc0 = __builtin_amdgcn_wmma_f32_16x16x4_f32(
                       false, qa[2 * c], false, kf0, (short)0, acc0, false, false);
            acc1 = __builtin_amdgcn_wmma_f32_16x16x4_f32(
                       false, qa[2 * c + 1], false, kf1, (short)0, acc1, false, false);
        }
        v8f acc = acc0 + acc1;
        const int* mk = mb + (size_t)(q0 + 8 * half) * SEQ + n0 + l16;
        unsigned myb = 0;
        #pragma unroll
        for (int i = 0; i < 8; ++i) {
            int   mv = mk[(size_t)i * SEQ];
            // ballot over all 32 lanes: bits[15:0] = row i, bits[31:16] = row i+8
            unsigned bb = __builtin_amdgcn_ballot_w32(mv != 0);
            myb = (lane == i) ? bb : myb;
            float s  = acc[i] * SM_SCALE + (float)mv * MASK_NEG;
            float mn = fmaxf(mrow[i], s);
            lrow[i]  = lrow[i] * __expf(mrow[i] - mn) + __expf(s - mn);
            mrow[i]  = mn;
        }
        if (lane < 8) mwave[(n0 >> 4) * 8 + lane] = myb;
    }

    __builtin_amdgcn_wave_barrier();
    asm volatile("s_wait_dscnt 0x0" ::: "memory");   // mask bits visible before pass 2
    __builtin_amdgcn_wave_barrier();

    // ---- Merge (m,l) across the 16 lanes of each half (rows live across lanes)
    float rln[8];
    #pragma unroll
    for (int i = 0; i < 8; ++i) {
        float m = mrow[i], l = lrow[i];
        #pragma unroll
        for (int off = 1; off < 16; off <<= 1) {
            float mo = __shfl_xor(m, off, 16);
            float lo = __shfl_xor(l, off, 16);
            float mn = fmaxf(m, mo);
            l = l * __expf(m - mn) + lo * __expf(mo - mn);
            m = mn;
        }
        mrow[i] = m;
        rln[i]  = 1.0f / l;
    }

    // ---- Pass 2: recompute scores, emit normalized P once (NT store), P@V via WMMA
    v8f ctx[4] = {{}, {}, {}, {}};
    float* myLds = plds[wav];

    for (int n0 = 0; n0 < SEQ; n0 += 16) {
        const float* krow = kb + (n0 + l16) * DH + 2 * half;
        __builtin_prefetch(krow + 16 * DH, 0, 1);
        v8f acc0 = {}, acc1 = {};
        #pragma unroll
        for (int c = 0; c < 8; ++c) {
            v2f kf0 = *(const v2f*)(krow + 4 * (2 * c));
            v2f kf1 = *(const v2f*)(krow + 4 * (2 * c + 1));
            acc0 = __builtin_amdgcn_wmma_f32_16x16x4_f32(
                       false, qa[2 * c], false, kf0, (short)0, acc0, false, false);
            acc1 = __builtin_amdgcn_wmma_f32_16x16x4_f32(
                       false, qa[2 * c + 1], false, kf1, (short)0, acc1, false, false);
        }
        v8f acc = acc0 + acc1;
        const unsigned* mbits = mwave + (n0 >> 4) * 8;
        float* arow = attb + (size_t)(q0 + 8 * half) * SEQ + n0 + l16;
        #pragma unroll
        for (int i = 0; i < 8; ++i) {
            float bf = (float)((mbits[i] >> (l16 + 16 * half)) & 1u);  // LDS broadcast read
            float s  = acc[i] * SM_SCALE + bf * MASK_NEG;              // identical math to pass 1
            float p  = __expf(s - mrow[i]) * rln[i];
            __builtin_nontemporal_store(p, arow + (size_t)i * SEQ);    // 1GB stream: keep out of L2
            myLds[(i + 8 * half) * 17 + l16] = p;                      // stage for transpose
        }
        __builtin_amdgcn_wave_barrier();
        asm volatile("s_wait_dscnt 0x0" ::: "memory");                 // intra-wave LDS visibility
        __builtin_amdgcn_wave_barrier();

        #pragma unroll
        for (int c = 0; c < 4; ++c) {
            v2f pa;                                                    // P in A-fragment layout
            pa.x = myLds[l16 * 17 + 4 * c + 2 * half];
            pa.y = myLds[l16 * 17 + 4 * c + 2 * half + 1];
            const float* vrow = vb + (size_t)(n0 + 4 * c + 2 * half) * DH + l16;
            #pragma unroll
            for (int j = 0; j < 4; ++j) {                              // 4 independent chains
                v2f vf;                                                // V B-fragment (K x N)
                vf.x = vrow[16 * j];
                vf.y = vrow[16 * j + DH];
                ctx[j] = __builtin_amdgcn_wmma_f32_16x16x4_f32(
                             false, pa, false, vf, (short)0, ctx[j], false, false);
            }
        }
        __builtin_amdgcn_wave_barrier();
        asm volatile("s_wait_dscnt 0x0" ::: "memory");                 // LDS reads done before reuse
        __builtin_amdgcn_wave_barrier();
    }

    // ---- Write context tile
    #pragma unroll
    for (int j = 0; j < 4; ++j)
        #pragma unroll
        for (int i = 0; i < 8; ++i)
            ctxb[(size_t)(q0 + i + 8 * half) * DH + 16 * j + l16] = ctx[j][i];
}

extern "C" void kernel_launch(void* const* d_in, const int* in_sizes, int n_in,
                              void* d_out, int out_size, void* d_ws, size_t ws_size,
                              hipStream_t stream) {
    const float* q    = (const float*)d_in[0];
    const float* k    = (const float*)d_in[1];
    const float* v    = (const float*)d_in[2];
    const int*   mask = (const int*)d_in[3];

    float* ctx = (float*)d_out;                                      // [B,H,S,D] first
    float* att = (float*)d_out + (size_t)NBATCH * NH * SEQ * DH;     // then [B,H,S,S]

    const int total_waves = NBATCH * NH * (SEQ / 16);                // 4096
    dim3 grid(total_waves / 8);                                      // 8 waves / 256-thread block
    dim3 block(256);
    attn_fused_kernel<<<grid, block, 0, stream>>>(q, k, v, mask, ctx, att);
}